// Lossv3_39256001085462
// MI455X (gfx1250) — compile-verified
//
#include <hip/hip_runtime.h>

// Lossv3: light-field warped masked-L1 loss.
// Stage 1: gather/compute kernel, one thread per (stack, batch, pixel).
//          Deterministic block-level LDS tree reduction -> partials in d_ws.
// Stage 2: single-block finalize; 16384 partials reduced in fixed order, with
//          the last 256->1 step done via V_WMMA_F32_16X16X4_F32 (ones x data:
//          sum of all D entries / 16 == sum of B entries, layout-agnostic).

#define HH   512
#define WW   512
#define BB   4
#define CC   3
#define TILE 16

typedef __attribute__((ext_vector_type(2))) float v2f;
typedef __attribute__((ext_vector_type(8))) float v8f;

__global__ __launch_bounds__(256) void warp_loss_kernel(
    const float* __restrict__ disp_in,
    const float* __restrict__ vx,
    const float* __restrict__ vy,
    const float* __restrict__ v45,
    const float* __restrict__ v135,
    float* __restrict__ partials)
{
  const int x  = blockIdx.x * TILE + threadIdx.x;
  const int y  = blockIdx.y * TILE + threadIdx.y;
  const int zb = blockIdx.z;          // zb = s*BB + b  (stack slowest -> L2 reuse)
  const int b  = zb & (BB - 1);
  const int s  = zb >> 2;

  const float* __restrict__ V =
      (s == 0) ? vx : (s == 1) ? vy : (s == 2) ? v45 : v135;
  // shift direction per stack: gx = x - dvm*t, gy = y - dum*t, t = (j-4)*disp
  const float dvm = (s == 0) ? 1.f : (s == 1) ? 0.f : (s == 2) ? 1.f : -1.f;
  const float dum = (s == 0) ? 0.f : 1.f;

  const int plane   = HH * WW;
  const int jstride = BB * CC * plane;           // views shape (9,B,C,H,W)
  const float disp  = disp_in[b * plane + y * WW + x];
  const float* __restrict__ vb = V + b * (CC * plane);

  float w[9][3];
#pragma unroll
  for (int j = 0; j < 9; ++j) {
    const float* __restrict__ pj = vb + j * jstride;
    if (j == 4) {                                // center view: identity warp
      const int o = y * WW + x;
      w[4][0] = pj[o];
      w[4][1] = pj[plane + o];
      w[4][2] = pj[2 * plane + o];
    } else {
      const float t  = (float)(j - 4) * disp;
      const float gx = fminf(fmaxf((float)x - dvm * t, 0.f), (float)(WW - 1));
      const float gy = fminf(fmaxf((float)y - dum * t, 0.f), (float)(HH - 1));
      const float x0f = floorf(gx), y0f = floorf(gy);
      const float wx = gx - x0f, wy = gy - y0f;
      int x0 = (int)x0f; x0 = min(max(x0, 0), WW - 1);
      int y0 = (int)y0f; y0 = min(max(y0, 0), HH - 1);
      const int x1 = min(x0 + 1, WW - 1);
      const int y1 = min(y0 + 1, HH - 1);
      const int o00 = y0 * WW + x0, o01 = y0 * WW + x1;
      const int o10 = y1 * WW + x0, o11 = y1 * WW + x1;
#pragma unroll
      for (int c = 0; c < 3; ++c) {
        const float* __restrict__ p = pj + c * plane;
        const float a = p[o00], bq = p[o01], cq = p[o10], d = p[o11];
        const float top = a  + wx * (bq - a);
        const float bot = cq + wx * (d  - cq);
        w[j][c] = top + wy * (bot - top);
      }
    }
  }

  // luma-weighted edge maps -> occlusion masks
  const float map_up =
      0.299f * fabsf(w[0][0] - w[4][0]) +
      0.587f * fabsf(w[0][1] - w[4][1]) +
      0.114f * fabsf(w[0][2] - w[4][2]);
  const float map_down =
      0.299f * fabsf(w[8][0] - w[4][0]) +
      0.587f * fabsf(w[8][1] - w[4][1]) +
      0.114f * fabsf(w[8][2] - w[4][2]);
  const float mask_up   = (map_up - map_down < 0.1f) ? 1.f : 0.f;
  const float mask_down = (map_down - map_up < 0.1f) ? 1.f : 0.f;

  float su = 0.f, sd = 0.f;
#pragma unroll
  for (int j = 0; j < 4; ++j)
#pragma unroll
    for (int c = 0; c < 3; ++c) su += fabsf(w[j][c] - w[4][c]);
#pragma unroll
  for (int j = 5; j < 9; ++j)
#pragma unroll
    for (int c = 0; c < 3; ++c) sd += fabsf(w[j][c] - w[4][c]);

  float part = mask_up * su + mask_down * sd;

  // deterministic block reduction
  __shared__ float red[256];
  const int tid = threadIdx.y * TILE + threadIdx.x;
  red[tid] = part;
  __syncthreads();
#pragma unroll
  for (int h = 128; h > 0; h >>= 1) {
    if (tid < h) red[tid] += red[tid + h];
    __syncthreads();
  }
  if (tid == 0) {
    const int bid = (blockIdx.z * gridDim.y + blockIdx.y) * gridDim.x + blockIdx.x;
    partials[bid] = red[0];
  }
}

__global__ __launch_bounds__(256) void finalize_kernel(
    const float* __restrict__ partials, int n, float* __restrict__ out)
{
  __shared__ float lds[256];
  __shared__ float lane_sums[32];
  const int tid = threadIdx.x;

  float s = 0.f;
  for (int i = tid; i < n; i += 256) s += partials[i];   // fixed order per lane
  lds[tid] = s;
  __syncthreads();

  if (tid < 32) {   // wave 0, EXEC all-1s inside the wave
#if __has_builtin(__builtin_amdgcn_wmma_f32_16x16x4_f32)
    // D = ones(16x4) x B(4x16) + C : every D[m][n] = column-sum of B, so
    // sum(D)/16 == sum(B) independent of the physical B register layout.
    v2f a; a[0] = 1.f; a[1] = 1.f;
    v8f c = {0.f, 0.f, 0.f, 0.f, 0.f, 0.f, 0.f, 0.f};
#pragma unroll
    for (int i = 0; i < 4; ++i) {
      v2f bm;
      bm[0] = lds[i * 64 + tid * 2 + 0];
      bm[1] = lds[i * 64 + tid * 2 + 1];
      c = __builtin_amdgcn_wmma_f32_16x16x4_f32(
          /*neg_a=*/false, a, /*neg_b=*/false, bm,
          /*c_mod=*/(short)0, c, /*reuse_a=*/false, /*reuse_b=*/false);
    }
    float ls = 0.f;
#pragma unroll
    for (int e = 0; e < 8; ++e) ls += c[e];
    lane_sums[tid] = ls * (1.0f / 16.0f);
#else
    float ls = 0.f;
#pragma unroll
    for (int e = 0; e < 8; ++e) ls += lds[tid * 8 + e];
    lane_sums[tid] = ls;
#endif
  }
  __syncthreads();

  if (tid == 0) {
    float t = 0.f;
#pragma unroll
    for (int i = 0; i < 32; ++i) t += lane_sums[i];
    // mean over (B,C,H,W) = 4*3*512*512 = 3145728
    *out = t * (1.0f / 3145728.0f);
  }
}

extern "C" void kernel_launch(void* const* d_in, const int* in_sizes, int n_in,
                              void* d_out, int out_size, void* d_ws, size_t ws_size,
                              hipStream_t stream) {
  const float* disp = (const float*)d_in[0];
  const float* vx   = (const float*)d_in[1];
  const float* vy   = (const float*)d_in[2];
  const float* v45  = (const float*)d_in[3];
  const float* v135 = (const float*)d_in[4];
  float* out        = (float*)d_out;
  float* partials   = (float*)d_ws;      // 16384 floats = 64 KB scratch

  dim3 block(TILE, TILE);
  dim3 grid(WW / TILE, HH / TILE, 4 * BB);   // 32 x 32 x 16 = 16384 blocks
  warp_loss_kernel<<<grid, block, 0, stream>>>(disp, vx, vy, v45, v135, partials);

  const int nPartials = (WW / TILE) * (HH / TILE) * 4 * BB;
  finalize_kernel<<<1, 256, 0, stream>>>(partials, nPartials, out);
}